// v_PTC_82403242541742
// MI455X (gfx1250) — compile-verified
//
#include <hip/hip_runtime.h>

typedef __attribute__((ext_vector_type(2))) float v2f;
typedef __attribute__((ext_vector_type(8))) float v8f;

#define FIN   4
#define FOUT  4
#define NPATH 9
#define TT    32
#define XX    16
#define YY    16
#define ZZ    16
#define VOL   (TT*XX*YY*ZZ)   // 131072
#define NS    4
#define NC    3
#define VOL96 ((unsigned)VOL * 96u)   // byte stride of one feature-channel i

// LDS arena (dynamic LDS; asm offsets = generic pointer low 32 bits, which
// are the LDS byte offset per ISA 10.2 aperture rules).
#define OFF_F0   0u          // float2[16][16][3] feature stage, buf 0 : 6144 B
#define OFF_U0   6144u       // link stage, buf 0 (1152 B + pad)      : 1536 B
#define OFF_F1   7680u       // feature stage, buf 1                  : 6144 B
#define OFF_U1   13824u      // link stage, buf 1                     : 1536 B
#define OFF_TR   15360u      // panel: 8 k-pair slabs x (96 R + 96 I) : 6144 B
#define LDS_TOTAL 21504u

// 12 feature B128 ops + 3 link B128 ops per path batch (wave-wide ASYNCcnt)
#define ASYNC_OPS_PER_PATH 15

// INST_OFFSET is added to BOTH the LDS and global addresses of async loads
// (ISA 08 §4.4); our LDS chunk layout mirrors global, so one immediate walks
// both sides in lockstep.
#define ASYNC_B128(lds, voff, saddr, ioff)                                   \
    asm volatile("global_load_async_to_lds_b128 %0, %1, %2 offset:" #ioff    \
                 :: "v"(lds), "v"(voff), "s"(saddr) : "memory")

// Complex MAC t += U*F with two V_PK_FMA_F32 (VOP3P op_sel broadcast + neg):
//  t.lo += U.x*F.x ; t.hi += U.x*F.y
//  t.lo += -U.y*F.y; t.hi += U.y*F.x
__device__ __forceinline__ void cmac_pk(v2f& t, v2f U, v2f F) {
    asm("v_pk_fma_f32 %0, %1, %2, %0 op_sel:[0,0,0] op_sel_hi:[0,1,1]"
        : "+v"(t) : "v"(U), "v"(F));
    asm("v_pk_fma_f32 %0, %1, %2, %0 op_sel:[1,1,0] op_sel_hi:[1,0,1] neg_lo:[0,1,0]"
        : "+v"(t) : "v"(U), "v"(F));
}

// One wave32 per 16-site z-line tile. Double-buffered ASYNCcnt staging +
// packed-FMA color rotation + Gauss 3-mult fp32 WMMA complex GEMM.
__global__ __launch_bounds__(32) void ptc_wmma_kernel(
    const float2* __restrict__ feat,    // [FIN][VOL][NS][NC] complex
    const float2* __restrict__ wgt,     // [FIN][FOUT][NPATH][NS][NS] complex
    const float2* __restrict__ trans,   // [NPATH][VOL][NC][NC] complex
    const int*    __restrict__ shifts,  // [NPATH][4]
    float2*       __restrict__ out)     // [FOUT][VOL][NS][NC] complex
{
    extern __shared__ char smem[];
    const unsigned smem_lo = (unsigned)(uintptr_t)smem;   // LDS byte offset

    const int lane = threadIdx.x;       // 0..31
    const int half = lane >> 4;
    const int l15  = lane & 15;

    const int tile   = blockIdx.x;
    const int t0     = tile / (XX*YY);
    const int x0     = (tile / YY) % XX;
    const int y0     = tile % YY;
    const int s_base = tile * 16;

    // ---- per-lane staging invariants (features: lane = site*2 + i-half) ----
    const int      site_ln = lane >> 1;
    const unsigned hv      = (lane & 1) ? 2u*VOL96 : 0u;     // i upper pair
    const unsigned lf0     = smem_lo + (unsigned)lane*192u;  // = site*384 + h*192
    const unsigned lane16  = (unsigned)lane*16u;
    const unsigned gv2     = (lane16 + 1024u > 1136u) ? 1136u
                                                      : (lane16 + 1024u); // clamped tail

    // Issue one path's async staging batch (exactly 15 wave-wide ops).
    auto stage = [&](int p, unsigned fbase, unsigned ubase) {
        const int st = shifts[p*4+0];
        const int sx = shifts[p*4+1];
        const int sy = shifts[p*4+2];
        const int sz = shifts[p*4+3];
        const int ts = (t0 - st + TT) % TT;
        const int xs = (x0 - sx + XX) % XX;
        const int ys = (y0 - sy + YY) % YY;
        const int src_base = ((ts*XX + xs)*YY + ys)*ZZ;      // + shifted z

        const unsigned long long sf =
            (unsigned long long)(uintptr_t)feat + (unsigned long long)src_base*96ull;
        const unsigned long long su =
            (unsigned long long)(uintptr_t)trans +
            ((unsigned long long)(unsigned)p*VOL + (unsigned)s_base)*72ull;

        // features: per lane a (site, i-pair); 12 x B128, 96 B per (site,i)
        const int      zs  = ((site_ln - sz) + ZZ) & (ZZ - 1);
        const unsigned vb0 = hv + (unsigned)zs*96u;
        const unsigned vb1 = vb0 + VOL96;
        const unsigned lb0 = lf0 + fbase;
        const unsigned lb1 = lb0 + 96u;
        ASYNC_B128(lb0, vb0, sf, 0);   ASYNC_B128(lb1, vb1, sf, 0);
        ASYNC_B128(lb0, vb0, sf, 16);  ASYNC_B128(lb1, vb1, sf, 16);
        ASYNC_B128(lb0, vb0, sf, 32);  ASYNC_B128(lb1, vb1, sf, 32);
        ASYNC_B128(lb0, vb0, sf, 48);  ASYNC_B128(lb1, vb1, sf, 48);
        ASYNC_B128(lb0, vb0, sf, 64);  ASYNC_B128(lb1, vb1, sf, 64);
        ASYNC_B128(lb0, vb0, sf, 80);  ASYNC_B128(lb1, vb1, sf, 80);

        // links: 1152 contiguous bytes -> 3 x B128 (tail clamped into pad)
        const unsigned lu = smem_lo + ubase + lane16;
        ASYNC_B128(lu,          lane16, su, 0);
        ASYNC_B128(lu,          lane16, su, 512);
        ASYNC_B128(lu + 1024u,  gv2,    su, 0);
    };

    // Gauss accumulators: P1 = sum AR*BR, P2 = sum AI*BI, P3 = sum ARI*BRI
    v8f P1[3] = {v8f{}, v8f{}, v8f{}};
    v8f P2[3] = {v8f{}, v8f{}, v8f{}};
    v8f P3[3] = {v8f{}, v8f{}, v8f{}};

    stage(0, OFF_F0, OFF_U0);   // prologue

    for (int p = 0; p < NPATH; ++p) {
        const int buf = p & 1;

        if (p + 1 < NPATH) {
            stage(p + 1, buf ? OFF_F0 : OFF_F1, buf ? OFF_U0 : OFF_U1);
            asm volatile("s_wait_asynccnt 15" ::: "memory");  // retire batch p
        } else {
            asm volatile("s_wait_asynccnt 0" ::: "memory");
        }
        __syncthreads();

        const v2f* fb = (const v2f*)(smem + (buf ? OFF_F1 : OFF_F0)); // [site][k][v]
        const v2f* ub = (const v2f*)(smem + (buf ? OFF_U1 : OFF_U0)); // [site*9+c*3+v]
        float*     pan = (float*)(smem + OFF_TR); // [q=k>>1][R:96 | I:96] floats

        // ---- color rotation -> panel; 2 x v_pk_fma_f32 per complex MAC ----
        // e = n*16 + k, e = lane + 32*it  =>  k = lane&15, n = half + 2*it.
        {
            const int k   = l15;
            const int qb  = (k >> 1)*192 + (k & 1);  // slab base + parity
            #pragma unroll 4
            for (int it = 0; it < 24; ++it) {
                const int n    = half + 2*it;
                const int site = n / 3;
                const int c    = n - site*3;
                v2f t = {0.f, 0.f};
                #pragma unroll
                for (int v = 0; v < 3; ++v)
                    cmac_pk(t, ub[site*9 + c*3 + v], fb[(site*16 + k)*3 + v]);
                pan[qb + n*2]      = t.x;   // R slab   (merges into one
                pan[qb + n*2 + 96] = t.y;   // I slab    ds_store_2addr_b32)
            }
        }

        // ---- A fragments: 16x16 complex W for path p, K-chunked by 4 ----
        v2f AR[4], AI[4], ARI[4];
        const int row = l15;                // M = o*4 + a
        const int o   = row >> 2, a = row & 3;
        #pragma unroll
        for (int kk = 0; kk < 4; ++kk) {
            #pragma unroll
            for (int j = 0; j < 2; ++j) {
                const int k = kk*4 + j + half*2;
                const int i = k >> 2, b = k & 3;
                const long idx = ((((long)i*FOUT + o)*NPATH + p)*NS + a)*NS + b;
                const float2 w = wgt[idx];
                AR[kk][j]  = w.x;
                AI[kk][j]  = w.y;
                ARI[kk][j] = w.x + w.y;
            }
        }
        __syncthreads();

        // ---- Gauss complex GEMM: 3 n-tiles x 4 K-chunks x 3 real WMMAs ----
        #pragma unroll
        for (int nt = 0; nt < 3; ++nt) {
            const int n = nt*16 + l15;
            #pragma unroll
            for (int kk = 0; kk < 4; ++kk) {
                const int qs = (kk*2 + half)*192;   // k-pair slab for (kb,kb+1)
                const v2f BR  = *(const v2f*)&pan[qs + n*2];       // one b64 load
                const v2f BI  = *(const v2f*)&pan[qs + n*2 + 96];  // one b64 load
                const v2f BRI = BR + BI;
                P1[nt] = __builtin_amdgcn_wmma_f32_16x16x4_f32(
                    false, AR[kk],  false, BR,  (short)0, P1[nt], false, false);
                P2[nt] = __builtin_amdgcn_wmma_f32_16x16x4_f32(
                    false, AI[kk],  false, BI,  (short)0, P2[nt], false, false);
                P3[nt] = __builtin_amdgcn_wmma_f32_16x16x4_f32(
                    false, ARI[kk], false, BRI, (short)0, P3[nt], false, false);
            }
        }
        __syncthreads();   // panel & stage buffers reused next path
    }

    // ---- reconstruct Re/Im and scatter D (VGPR r -> M = r + 8*half, N = l15) ----
    #pragma unroll
    for (int nt = 0; nt < 3; ++nt) {
        const int n    = nt*16 + l15;
        const int site = s_base + n/3;
        const int c    = n % 3;
        #pragma unroll
        for (int r = 0; r < 8; ++r) {
            const int M = r + half*8;
            const int o = M >> 2, a = M & 3;
            const float re = P1[nt][r] - P2[nt][r];
            const float im = P3[nt][r] - P1[nt][r] - P2[nt][r];
            const long idx = (((long)o*VOL + site)*NS + a)*NC + c;
            out[idx] = make_float2(re, im);
        }
    }
}

extern "C" void kernel_launch(void* const* d_in, const int* in_sizes, int n_in,
                              void* d_out, int out_size, void* d_ws, size_t ws_size,
                              hipStream_t stream) {
    const float2* feat   = (const float2*)d_in[0];
    const float2* wgt    = (const float2*)d_in[1];
    const float2* trans  = (const float2*)d_in[2];
    const int*    shifts = (const int*)d_in[3];
    float2* out = (float2*)d_out;

    dim3 grid(VOL / 16);   // 8192 z-line tiles
    dim3 block(32);        // one wave32 per tile
    hipLaunchKernelGGL(ptc_wmma_kernel, grid, block, LDS_TOTAL, stream,
                       feat, wgt, trans, shifts, out);
}